// FactorizationMachineModel_70342974374395
// MI455X (gfx1250) — compile-verified
//
#include <hip/hip_runtime.h>
#include <stdint.h>

// ---- types for CDNA5 intrinsics ----
typedef __attribute__((ext_vector_type(2))) float        v2f;
typedef __attribute__((ext_vector_type(8))) float        v8f;
typedef __attribute__((ext_vector_type(4))) unsigned int u32x4;
typedef __attribute__((ext_vector_type(8))) int          i32x8;
typedef __attribute__((ext_vector_type(4))) int          i32x4;

#define B_ROWS      1024
#define F_DIM       100000
#define D_DIM       16
#define KT          800            // K-tile per workgroup (125 * 800 = 100000)
#define ROWS_PER_WG 16
#define NWAVES      8
#define COLS_PER_WAVE (KT / NWAVES)    // 100
#define NSTEPS        (COLS_PER_WAVE / 4) // 25 WMMA K=4 steps per wave

// ---------------- zero workspace ----------------
__global__ __launch_bounds__(256) void fm_zero(float* ws, int n) {
    int i = blockIdx.x * blockDim.x + threadIdx.x;
    if (i < n) ws[i] = 0.0f;
}

// ---------------- main GEMM kernel ----------------
// grid = (64 row-blocks, 125 K-chunks), block = 256 (8 waves)
__global__ __launch_bounds__(256) void fm_main(const float* __restrict__ x,
                                               const float* __restrict__ V,
                                               const float* __restrict__ W,
                                               float* __restrict__ ws_xv,   // [1024][16]
                                               float* __restrict__ ws_q,    // [1024]
                                               float* __restrict__ ws_lin)  // [1024]
{
    __shared__ float tile[ROWS_PER_WG * KT];   // 51200 B: x tile, row-major
    __shared__ float lds_xv[ROWS_PER_WG * 16];
    __shared__ float lds_q[ROWS_PER_WG];
    __shared__ float lds_lin[ROWS_PER_WG];

    const int rb   = blockIdx.x;           // row block (16 rows)
    const int k0   = blockIdx.y * KT;      // K chunk start
    const int tid  = threadIdx.x;
    const int wave = tid >> 5;
    const int lane = tid & 31;
    const int m    = lane & 15;            // A row / B col / C col within tile
    const int half = lane >> 4;            // 0: lanes 0-15, 1: lanes 16-31
    const int koff = half * 2;             // K-pair offset within a K=4 step

    // zero LDS reduction buffers (barrier below covers this)
    lds_xv[tid] = 0.0f;
    if (tid < ROWS_PER_WG) { lds_q[tid] = 0.0f; lds_lin[tid] = 0.0f; }

    // -------- TDM: DMA the 16xKT fp32 x-tile into LDS (wave 0 only) --------
    if (wave == 0) {
        const uint64_t ga = (uint64_t)(uintptr_t)(x + (size_t)rb * ROWS_PER_WG * F_DIM + k0);
        const uint32_t lds_off = (uint32_t)(uintptr_t)(&tile[0]);

        u32x4 g0;
        g0[0] = 1u;                                   // count=1, user descriptor
        g0[1] = lds_off;                              // lds_addr [63:32]
        g0[2] = (uint32_t)ga;                         // global_addr[31:0]
        g0[3] = (uint32_t)((ga >> 32) & 0x1FFFFFFu)   // global_addr[56:32]
              | (2u << 30);                           // type = 2 ("image")

        i32x8 g1;
        g1[0] = (int)(2u << 16);                      // wg_mask=0, data_size=2 (4 bytes)
        g1[1] = (int)(((uint32_t)F_DIM & 0xFFFFu) << 16);          // tensor_dim0[15:0]
        g1[2] = (int)(((uint32_t)F_DIM >> 16) & 0xFFFFu)           // tensor_dim0[31:16]
              | (int)(((uint32_t)B_ROWS & 0xFFFFu) << 16);         // tensor_dim1[15:0]
        g1[3] = (int)(((uint32_t)B_ROWS >> 16) & 0xFFFFu)          // tensor_dim1[31:16]
              | (int)((uint32_t)KT << 16);                         // tile_dim0 = KT
        g1[4] = (int)(ROWS_PER_WG & 0xFFFF);                       // tile_dim1=16, tile_dim2=0
        g1[5] = (int)F_DIM;                                        // tensor_dim0_stride[31:0]
        g1[6] = 0;                                                 // stride0 hi, dim1_stride lo
        g1[7] = 0;

        i32x4 gz4 = {0, 0, 0, 0};                      // groups 2/3: dims beyond 2D unused
        i32x8 gz8 = {0, 0, 0, 0, 0, 0, 0, 0};          // extra group (clang-23 6-arg form)
        __builtin_amdgcn_tensor_load_to_lds(g0, g1, gz4, gz4, gz8, 0);
        __builtin_amdgcn_s_wait_tensorcnt(0);
    }
    __syncthreads();

    // -------- WMMA accumulation over this wave's K columns --------
    v8f acc1 = {0.f,0.f,0.f,0.f,0.f,0.f,0.f,0.f};  // x @ V        (16x16 tile)
    v8f acc2 = {0.f,0.f,0.f,0.f,0.f,0.f,0.f,0.f};  // x^2 @ V^2    (16x16 tile)
    float lin = 0.0f;

    const int cbase = wave * COLS_PER_WAVE;
    #pragma unroll 5
    for (int s = 0; s < NSTEPS; ++s) {
        const int kc = cbase + s * 4;

        // A (16x4 fp32): lane m holds row m, K pair {koff, koff+1}  -> ds_load_b64
        v2f a = *(const v2f*)(&tile[m * KT + kc + koff]);

        // B (4x16 fp32): lane n=m holds col n, same K-pair striping as C layout
        const int kg = k0 + kc + koff;
        v2f b;
        b[0] = V[(size_t)kg       * D_DIM + m];
        b[1] = V[(size_t)(kg + 1) * D_DIM + m];

        // linear term: W is contiguous in K -> 8B load
        v2f w = *(const v2f*)(&W[kg]);
        lin = __builtin_fmaf(a[0], w[0], lin);
        lin = __builtin_fmaf(a[1], w[1], lin);

        v2f a2 = a * a;
        v2f b2 = b * b;

        acc1 = __builtin_amdgcn_wmma_f32_16x16x4_f32(false, a,  false, b,  (short)0, acc1, false, false);
        acc2 = __builtin_amdgcn_wmma_f32_16x16x4_f32(false, a2, false, b2, (short)0, acc2, false, false);
    }

    // -------- reductions --------
    // q[row] = sum over N of acc2 tile: shfl-reduce across the 16 N-lanes per VGPR
    #pragma unroll
    for (int i = 0; i < 8; ++i) {
        float v = acc2[i];
        v += __shfl_xor(v, 1, 32);
        v += __shfl_xor(v, 2, 32);
        v += __shfl_xor(v, 4, 32);
        v += __shfl_xor(v, 8, 32);
        if (m == 0) atomicAdd(&lds_q[i + 8 * half], v);   // row = i + 8*half
    }
    // lin: lanes L and L+16 hold disjoint K pieces of the same row m
    atomicAdd(&lds_lin[m], lin);
    // xv tile: C layout -> (row = i + 8*half, col = m)
    #pragma unroll
    for (int i = 0; i < 8; ++i)
        atomicAdd(&lds_xv[(i + 8 * half) * 16 + m], acc1[i]);

    __syncthreads();

    // -------- flush per-WG partials to global workspace --------
    atomicAdd(&ws_xv[((size_t)rb * ROWS_PER_WG + (tid >> 4)) * 16 + (tid & 15)], lds_xv[tid]);
    if (tid < ROWS_PER_WG) {
        atomicAdd(&ws_q[rb * ROWS_PER_WG + tid],   lds_q[tid]);
        atomicAdd(&ws_lin[rb * ROWS_PER_WG + tid], lds_lin[tid]);
    }
}

// ---------------- final combine ----------------
__global__ __launch_bounds__(256) void fm_final(const float* __restrict__ ws_xv,
                                                const float* __restrict__ ws_q,
                                                const float* __restrict__ ws_lin,
                                                const float* __restrict__ b_lin,
                                                float* __restrict__ out)
{
    int b = blockIdx.x * blockDim.x + threadIdx.x;
    if (b < B_ROWS) {
        float s = 0.0f;
        #pragma unroll
        for (int d = 0; d < D_DIM; ++d) {
            float t = ws_xv[b * D_DIM + d];
            s = __builtin_fmaf(t, t, s);
        }
        out[b] = 0.5f * (s - ws_q[b]) + ws_lin[b] + b_lin[0];
    }
}

extern "C" void kernel_launch(void* const* d_in, const int* in_sizes, int n_in,
                              void* d_out, int out_size, void* d_ws, size_t ws_size,
                              hipStream_t stream) {
    const float* x  = (const float*)d_in[0];   // [1024, 100000]
    const float* V  = (const float*)d_in[1];   // [100000, 16]
    const float* W  = (const float*)d_in[2];   // [1, 100000]
    const float* bl = (const float*)d_in[3];   // [1]

    float* ws     = (float*)d_ws;
    float* ws_xv  = ws;                 // 1024*16 = 16384 floats
    float* ws_q   = ws + 16384;         // 1024 floats
    float* ws_lin = ws + 16384 + 1024;  // 1024 floats
    const int nz = 16384 + 1024 + 1024; // 18432 floats (73.7 KB of d_ws)

    fm_zero<<<(nz + 255) / 256, 256, 0, stream>>>(ws, nz);
    fm_main<<<dim3(B_ROWS / ROWS_PER_WG, F_DIM / KT), 256, 0, stream>>>(x, V, W, ws_xv, ws_q, ws_lin);
    fm_final<<<(B_ROWS + 255) / 256, 256, 0, stream>>>(ws_xv, ws_q, ws_lin, bl, (float*)d_out);
}